// NGCFLayer_our5_52561809769220
// MI455X (gfx1250) — compile-verified
//
#include <hip/hip_runtime.h>

// Problem constants (from reference)
#define NU 100000
#define NI 50000
#define NE 600000
#define D  128
#define LDSP 129          // LDS row pitch in floats (bank-conflict-free: 129 % 64 = 1)
#define NEG_SLOPE 0.2f
#define EPSN 1e-12f

typedef float v2f __attribute__((ext_vector_type(2)));
typedef float v8f __attribute__((ext_vector_type(8)));

// sched_group_barrier masks
#define SG_WMMA    0x008
#define SG_VMEMRD  0x020
#define SG_DSRD    0x100

__device__ __forceinline__ void atom_add_f32(float* p, float v) {
  (void)__hip_atomic_fetch_add(p, v, __ATOMIC_RELAXED, __HIP_MEMORY_SCOPE_AGENT);
}

// ---------------------------------------------------------------------------
// Kernel 1: initialize workspace accumulators.
//   xu = feat_user (so X_user = feat_user + agg_user accumulates in place)
//   xi = feat_item
//   mu = mi = 0
// ---------------------------------------------------------------------------
__global__ void __launch_bounds__(256) ngcf_init(
    const float* __restrict__ fu, const float* __restrict__ fi,
    float* __restrict__ xu, float* __restrict__ mu,
    float* __restrict__ xi, float* __restrict__ mi) {
  const long NUD = (long)NU * D;
  const long NID = (long)NI * D;
  const long stride = (long)gridDim.x * blockDim.x;
  long t0 = (long)blockIdx.x * blockDim.x + threadIdx.x;
  for (long t = t0; t < NUD; t += stride) { xu[t] = fu[t]; mu[t] = 0.0f; }
  for (long t = t0; t < NID; t += stride) { xi[t] = fi[t]; mi[t] = 0.0f; }
}

// ---------------------------------------------------------------------------
// Kernel 2: edge scatter phase. One wave32 per edge, 4 floats per lane.
//   xu[u] += feat_item[i] * norm_iu[e]        (agg_user)
//   xi[i] += feat_user[u] * norm_ui[e]        (agg_item)
//   msg   = feat_user[u]*norm_user[u] * feat_item[i]*norm_item[i]
//   mu[u] += msg ; mi[i] += msg
// ---------------------------------------------------------------------------
__global__ void __launch_bounds__(256) ngcf_edge(
    const float* __restrict__ fu, const float* __restrict__ fi,
    const float* __restrict__ nu, const float* __restrict__ ni,
    const float* __restrict__ nui, const float* __restrict__ niu,
    const int* __restrict__ eu, const int* __restrict__ ei,
    float* __restrict__ xu, float* __restrict__ mu,
    float* __restrict__ xi, float* __restrict__ mi) {
  const int e = (blockIdx.x * blockDim.x + threadIdx.x) >> 5;
  if (e >= NE) return;
  const int lane = threadIdx.x & 31;

  const int u = eu[e];
  const int i = ei[e];
  const float su  = nu[u];
  const float si  = ni[i];
  const float sui = nui[e];
  const float siu = niu[e];

  const float4 a = ((const float4*)(fu + (size_t)u * D))[lane];  // feat_user[u]
  const float4 b = ((const float4*)(fi + (size_t)i * D))[lane];  // feat_item[i]

  const float ms = su * si;
  float4 m;
  m.x = (a.x * b.x) * ms;
  m.y = (a.y * b.y) * ms;
  m.z = (a.z * b.z) * ms;
  m.w = (a.w * b.w) * ms;

  const size_t ou = (size_t)u * D + (size_t)lane * 4;
  const size_t oi = (size_t)i * D + (size_t)lane * 4;

  // agg_user += feat_item * norm_iu
  atom_add_f32(xu + ou + 0, b.x * siu);
  atom_add_f32(xu + ou + 1, b.y * siu);
  atom_add_f32(xu + ou + 2, b.z * siu);
  atom_add_f32(xu + ou + 3, b.w * siu);
  // agg_item += feat_user * norm_ui
  atom_add_f32(xi + oi + 0, a.x * sui);
  atom_add_f32(xi + oi + 1, a.y * sui);
  atom_add_f32(xi + oi + 2, a.z * sui);
  atom_add_f32(xi + oi + 3, a.w * sui);
  // msg_user += msg ; msg_item += msg
  atom_add_f32(mu + ou + 0, m.x);
  atom_add_f32(mu + ou + 1, m.y);
  atom_add_f32(mu + ou + 2, m.z);
  atom_add_f32(mu + ou + 3, m.w);
  atom_add_f32(mi + oi + 0, m.x);
  atom_add_f32(mi + oi + 1, m.y);
  atom_add_f32(mi + oi + 2, m.z);
  atom_add_f32(mi + oi + 3, m.w);
}

// ---------------------------------------------------------------------------
// Kernel 3: fused dual-GEMM + bias + leaky-relu + row-L2-normalize.
//   H = X @ W1^T + M @ W2^T + (b1 + b2)   ->  post()  ->  out
// Block = 256 threads = 8 waves; each wave owns 16 rows x 128 cols of output.
// W1/W2 staged in LDS (row pitch 129 floats -> conflict-free B reads).
// WMMA f32 16x16x4: A 16x4 (lanes 0-15: K{0,1}; lanes 16-31: K{2,3}),
// B 4x16 = W[n,k] slice, C/D 16x16 in 8 VGPRs.
// Per k-step pipeline shape is pinned with sched_group_barrier:
//   1 VMEM read (A) -> 8 DS reads (B fragments) -> 8 WMMA
// so DS waits amortize instead of a full s_wait_dscnt 0 before every WMMA.
// ---------------------------------------------------------------------------
__global__ void __launch_bounds__(256) ngcf_gemm(
    const float* __restrict__ X, const float* __restrict__ Mm,
    const float* __restrict__ W1, const float* __restrict__ W2,
    const float* __restrict__ b1, const float* __restrict__ b2,
    float* __restrict__ out, int nrows) {
  extern __shared__ float lds[];
  float* w1s = lds;                // [D][LDSP]
  float* w2s = lds + D * LDSP;     // [D][LDSP]

  const int tid = threadIdx.x;
  // Stage W1/W2 with b128 global reads (LDS stores stay scalar: the 129-float
  // pitch intentionally breaks 16B alignment to kill bank conflicts on reads).
  for (int idx = tid; idx < (D * D) / 4; idx += 256) {
    const int base = idx * 4;
    const int n = base >> 7;
    const int k = base & (D - 1);
    const float4 w1v = ((const float4*)W1)[idx];
    const float4 w2v = ((const float4*)W2)[idx];
    float* p1 = w1s + n * LDSP + k;
    float* p2 = w2s + n * LDSP + k;
    p1[0] = w1v.x; p1[1] = w1v.y; p1[2] = w1v.z; p1[3] = w1v.w;
    p2[0] = w2v.x; p2[1] = w2v.y; p2[2] = w2v.z; p2[3] = w2v.w;
  }
  __syncthreads();

  const int wave = tid >> 5;
  const int lane = tid & 31;
  const int ln   = lane & 15;      // position within half-wave
  const int hi   = lane >> 4;      // half-wave select
  const int row0 = blockIdx.x * 128 + wave * 16;

  const int r  = row0 + ln;
  const int rc = (r < nrows) ? r : (nrows - 1);   // clamp OOB rows (masked at store)

  v8f c[8] = {};                   // 8 N-tiles of 16x16 f32 accumulators

  const float2* ax = (const float2*)(X  + (size_t)rc * D);
  const float2* am = (const float2*)(Mm + (size_t)rc * D);

  // ---- X @ W1^T ----
#pragma unroll 2
  for (int kk = 0; kk < D / 4; ++kk) {
    const float2 av = ax[kk * 2 + hi];           // K = 4*kk + 2*hi + {0,1}
    v2f a; a.x = av.x; a.y = av.y;
    const float* wb = w1s + kk * 4 + 2 * hi;
    float bx[8], by[8];
#pragma unroll
    for (int j = 0; j < 8; ++j) {
      const float* wp = wb + (j * 16 + ln) * LDSP;  // B[k][n] = W1[n][k]
      bx[j] = wp[0];
      by[j] = wp[1];
    }
#pragma unroll
    for (int j = 0; j < 8; ++j) {
      v2f b; b.x = bx[j]; b.y = by[j];
      c[j] = __builtin_amdgcn_wmma_f32_16x16x4_f32(
          false, a, false, b, (short)0, c[j], false, false);
    }
    // Pin pipeline shape: A load, then all B loads, then all WMMAs.
    __builtin_amdgcn_sched_group_barrier(SG_VMEMRD, 1, 0);
    __builtin_amdgcn_sched_group_barrier(SG_DSRD,   8, 0);
    __builtin_amdgcn_sched_group_barrier(SG_WMMA,   8, 0);
  }
  // ---- + M @ W2^T ----
#pragma unroll 2
  for (int kk = 0; kk < D / 4; ++kk) {
    const float2 av = am[kk * 2 + hi];
    v2f a; a.x = av.x; a.y = av.y;
    const float* wb = w2s + kk * 4 + 2 * hi;
    float bx[8], by[8];
#pragma unroll
    for (int j = 0; j < 8; ++j) {
      const float* wp = wb + (j * 16 + ln) * LDSP;
      bx[j] = wp[0];
      by[j] = wp[1];
    }
#pragma unroll
    for (int j = 0; j < 8; ++j) {
      v2f b; b.x = bx[j]; b.y = by[j];
      c[j] = __builtin_amdgcn_wmma_f32_16x16x4_f32(
          false, a, false, b, (short)0, c[j], false, false);
    }
    __builtin_amdgcn_sched_group_barrier(SG_VMEMRD, 1, 0);
    __builtin_amdgcn_sched_group_barrier(SG_DSRD,   8, 0);
    __builtin_amdgcn_sched_group_barrier(SG_WMMA,   8, 0);
  }

  // ---- epilogue: bias + leaky-relu + row L2 norm ----
  float bb[8];
#pragma unroll
  for (int j = 0; j < 8; ++j) bb[j] = b1[j * 16 + ln] + b2[j * 16 + ln];

  float ss[8];
#pragma unroll
  for (int v = 0; v < 8; ++v) ss[v] = 0.0f;

  // C layout: VGPR v -> row (v + 8*hi), col (j*16 + ln)
#pragma unroll
  for (int j = 0; j < 8; ++j) {
#pragma unroll
    for (int v = 0; v < 8; ++v) {
      float h = c[j][v] + bb[j];
      h = (h > 0.0f) ? h : NEG_SLOPE * h;
      c[j][v] = h;
      ss[v] += h * h;
    }
  }

  // reduce sum-of-squares across the 16 lanes that share a row (xor < 16 stays
  // within each half-wave)
#pragma unroll
  for (int v = 0; v < 8; ++v) {
    float s = ss[v];
    s += __shfl_xor(s, 1, 32);
    s += __shfl_xor(s, 2, 32);
    s += __shfl_xor(s, 4, 32);
    s += __shfl_xor(s, 8, 32);
    ss[v] = 1.0f / fmaxf(sqrtf(s), EPSN);
  }

#pragma unroll
  for (int v = 0; v < 8; ++v) {
    const int row = row0 + v + 8 * hi;
    if (row < nrows) {
      float* op = out + (size_t)row * D + ln;
#pragma unroll
      for (int j = 0; j < 8; ++j) op[j * 16] = c[j][v] * ss[v];
    }
  }
}

// ---------------------------------------------------------------------------
// Host launcher
// ---------------------------------------------------------------------------
extern "C" void kernel_launch(void* const* d_in, const int* in_sizes, int n_in,
                              void* d_out, int out_size, void* d_ws, size_t ws_size,
                              hipStream_t stream) {
  const float* feat_user = (const float*)d_in[0];
  const float* feat_item = (const float*)d_in[1];
  const float* norm_user = (const float*)d_in[2];
  const float* norm_item = (const float*)d_in[3];
  const float* norm_ui   = (const float*)d_in[4];
  const float* norm_iu   = (const float*)d_in[5];
  const float* W1        = (const float*)d_in[6];
  const float* b1        = (const float*)d_in[7];
  const float* W2        = (const float*)d_in[8];
  const float* b2        = (const float*)d_in[9];
  const int*   edge_u    = (const int*)d_in[10];
  const int*   edge_i    = (const int*)d_in[11];

  float* xu = (float*)d_ws;                       // [NU, D]  feat_user + agg_user
  float* mu = xu + (size_t)NU * D;                // [NU, D]  msg_user
  float* xi = mu + (size_t)NU * D;                // [NI, D]  feat_item + agg_item
  float* mi = xi + (size_t)NI * D;                // [NI, D]  msg_item

  float* out_user = (float*)d_out;                // [NU, D]
  float* out_item = out_user + (size_t)NU * D;    // [NI, D]

  ngcf_init<<<2048, 256, 0, stream>>>(feat_user, feat_item, xu, mu, xi, mi);

  ngcf_edge<<<(NE + 7) / 8, 256, 0, stream>>>(
      feat_user, feat_item, norm_user, norm_item, norm_ui, norm_iu,
      edge_u, edge_i, xu, mu, xi, mi);

  const size_t ldsb = (size_t)2 * D * LDSP * sizeof(float);   // 132,096 B
  ngcf_gemm<<<(NU + 127) / 128, 256, ldsb, stream>>>(
      xu, mu, W1, W2, b1, b2, out_user, NU);
  ngcf_gemm<<<(NI + 127) / 128, 256, ldsb, stream>>>(
      xi, mi, W1, W2, b1, b2, out_item, NI);
}